// EMADecomposition_75952201662654
// MI455X (gfx1250) — compile-verified
//
#include <hip/hip_runtime.h>
#include <cstdint>
#include <cstddef>

// EMA decomposition (B,L,C)=(32,4096,512) f32.
// trend_t = a*trend_{t-1} + (1-a)*x_t, trend_0 = x_0, a = sigmoid(alpha[c])
// seasonal = x - trend. Output = [seasonal | trend] concatenated.
//
// 3-phase grid-level blocked scan over L, tiled so every wave accesses
// consecutive channels (perfect coalescing), with CDNA5 async global->LDS
// staging (global_load_async_to_lds_b128 + s_wait_asynccnt) in the fixup pass.

typedef float v2f __attribute__((ext_vector_type(2)));

#define Bn     32
#define Lt     4096
#define Ct     512
#define NSLAB  32
#define ROWS   128           // Lt / NSLAB rows per slab
#define HALF   64            // double-buffer half (rows)
#define TPB    256           // 8 wave32 per block; thread owns a channel pair
#define HALF_BYTES (HALF * Ct * 4)   // 131072 B per LDS half-buffer

__device__ __forceinline__ v2f sigmoid2(v2f v) {
  v2f r;
  r.x = 1.0f / (1.0f + __expf(-v.x));
  r.y = 1.0f / (1.0f + __expf(-v.y));
  return r;
}

// ---------------- Phase 1: per-slab local scan -> carry-out ----------------
__global__ void k_partial(const float* __restrict__ x,
                          const float* __restrict__ alpha,
                          float* __restrict__ carry) {
  const int tid = threadIdx.x;
  const int b   = blockIdx.x >> 5;          // / NSLAB
  const int s   = blockIdx.x & (NSLAB - 1);
  const int c   = 2 * tid;

  v2f a  = sigmoid2(*(const v2f*)(alpha + c));
  v2f bs; bs.x = 1.0f - a.x; bs.y = 1.0f - a.y;

  const float* base = x + ((size_t)b * Lt + (size_t)s * ROWS) * Ct + c;

  v2f y; y.x = 0.0f; y.y = 0.0f;
  #pragma unroll 4
  for (int l = 0; l < ROWS; ++l) {
    v2f xv = *(const v2f*)(base + (size_t)l * Ct);   // wave: 256B contiguous
    v2f yn = a * y + bs * xv;
    if (s == 0 && l == 0) yn = xv;                   // trend_0 = x_0 exactly
    y = yn;
  }
  *(v2f*)(carry + ((size_t)(b * NSLAB + s)) * Ct + c) = y;
}

// ---------------- Phase 2: combine carries across slabs --------------------
// cin(0)=unused(0); cin(1)=local(0); cin(s+1)=P*cin(s)+local(s), P=a^128.
__global__ void k_combine(const float* __restrict__ alpha,
                          const float* __restrict__ carry,
                          float* __restrict__ cin) {
  const int tid = threadIdx.x;
  const int b   = blockIdx.x;
  const int c   = 2 * tid;

  v2f a = sigmoid2(*(const v2f*)(alpha + c));
  v2f P = a;
  #pragma unroll
  for (int i = 0; i < 7; ++i) P = P * P;             // a^128

  v2f acc; acc.x = 0.0f; acc.y = 0.0f;
  for (int s = 0; s < NSLAB; ++s) {
    const size_t idx = ((size_t)(b * NSLAB + s)) * Ct + c;
    *(v2f*)(cin + idx) = acc;
    v2f loc = *(const v2f*)(carry + idx);
    if (s == 0) acc = loc;                           // slab0 local == true Y(0)
    else        acc = P * acc + loc;
  }
}

// ---------------- Phase 3: fixup re-scan with async LDS staging ------------
__global__ void k_fixup(const float* __restrict__ x,
                        const float* __restrict__ alpha,
                        const float* __restrict__ cin,
                        float* __restrict__ outS,
                        float* __restrict__ outT) {
  extern __shared__ float smem[];                    // 2 * HALF_BYTES dynamic
  const int tid = threadIdx.x;
  const int b   = blockIdx.x >> 5;
  const int s   = blockIdx.x & (NSLAB - 1);
  const int c   = 2 * tid;

  v2f a  = sigmoid2(*(const v2f*)(alpha + c));
  v2f bs; bs.x = 1.0f - a.x; bs.y = 1.0f - a.y;

  v2f y = *(const v2f*)(cin + ((size_t)(b * NSLAB + s)) * Ct + c);

  const float* slab = x + ((size_t)b * Lt + (size_t)s * ROWS) * Ct;
  const unsigned lds0 = (unsigned)(uintptr_t)smem;   // low 32 bits = LDS offset

  // Stage half 0: 128 KB, each thread issues 32x 16B async transfers,
  // lanes cover 512B contiguous per instruction round.
  {
    const char* g = (const char*)slab;
    #pragma unroll
    for (int i = 0; i < 32; ++i) {
      const unsigned off = (unsigned)((i * TPB + tid) * 16);
      const unsigned la  = lds0 + off;
      const char*    ga  = g + off;
      asm volatile("global_load_async_to_lds_b128 %0, %1, off"
                   :: "v"(la), "v"(ga) : "memory");
    }
  }
  asm volatile("s_wait_asynccnt 0x0" ::: "memory");
  __syncthreads();

  for (int h = 0; h < 2; ++h) {
    if (h == 0) {
      // Overlap: stage half 1 while scanning half 0.
      const char*    g     = (const char*)(slab + (size_t)HALF * Ct);
      const unsigned lbase = lds0 + (unsigned)HALF_BYTES;
      #pragma unroll
      for (int i = 0; i < 32; ++i) {
        const unsigned off = (unsigned)((i * TPB + tid) * 16);
        const unsigned la  = lbase + off;
        const char*    ga  = g + off;
        asm volatile("global_load_async_to_lds_b128 %0, %1, off"
                     :: "v"(la), "v"(ga) : "memory");
      }
    }

    const float* sm  = smem + (size_t)h * (HALF * Ct);
    const int    gl0 = s * ROWS + h * HALF;
    float* oS = outS + ((size_t)b * Lt + gl0) * Ct + c;
    float* oT = outT + ((size_t)b * Lt + gl0) * Ct + c;

    #pragma unroll 4
    for (int l = 0; l < HALF; ++l) {
      v2f xv = *(const v2f*)(sm + (size_t)l * Ct + c);   // ds_load_b64, no bank conflicts
      v2f yn = a * y + bs * xv;
      if ((gl0 + l) == 0) yn = xv;                       // boundary: trend_0 = x_0
      y = yn;
      v2f se = xv - y;
      // Non-temporal: 512MB of streaming output must not thrash L2-resident x.
      __builtin_nontemporal_store(se, (v2f*)(oS + (size_t)l * Ct));
      __builtin_nontemporal_store(y,  (v2f*)(oT + (size_t)l * Ct));
    }

    if (h == 0) {
      asm volatile("s_wait_asynccnt 0x0" ::: "memory");
      __syncthreads();
    }
  }
}

// ---------------------------------------------------------------------------
extern "C" void kernel_launch(void* const* d_in, const int* in_sizes, int n_in,
                              void* d_out, int out_size, void* d_ws, size_t ws_size,
                              hipStream_t stream) {
  (void)in_sizes; (void)n_in; (void)out_size; (void)ws_size;

  const float* x     = (const float*)d_in[0];
  const float* alpha = (const float*)d_in[1];

  float* outS = (float*)d_out;                          // seasonal first
  float* outT = outS + (size_t)Bn * Lt * Ct;            // then trend

  float* carry = (float*)d_ws;                          // Bn*NSLAB*Ct f32 (2MB)
  float* cin   = carry + (size_t)Bn * NSLAB * Ct;       // Bn*NSLAB*Ct f32 (2MB)

  k_partial<<<Bn * NSLAB, TPB, 0, stream>>>(x, alpha, carry);
  k_combine<<<Bn,          TPB, 0, stream>>>(alpha, carry, cin);
  k_fixup  <<<Bn * NSLAB, TPB, 2 * HALF_BYTES, stream>>>(x, alpha, cin, outS, outT);
}